// WhiteCat_28406913696447
// MI455X (gfx1250) — compile-verified
//
#include <hip/hip_runtime.h>
#include <stdint.h>

// ---------------------------------------------------------------------------
// Channel-wise codebook dequant (gather):
//   out[b,c,h,w] = codebook[c, cat(left,right)[b,c,h,w]]
// B=32, CL=CR=128, C=256, H=W=64, NUM_CODES=256.
// Pure bandwidth-bound gather: 128 MiB idx in + 128 MiB f32 out ~= 11.5us
// at 23.3 TB/s. Strategy: one block per (b,c) plane; stage the 1 KiB
// codebook row into LDS via the gfx1250 async global->LDS path, gather via
// LDS (ds_load_b32), stream idx/out with nontemporal B128 accesses.
// ---------------------------------------------------------------------------

typedef int   v4i __attribute__((ext_vector_type(4)));
typedef float v4f __attribute__((ext_vector_type(4)));

#define AS1 __attribute__((address_space(1)))
#define AS3 __attribute__((address_space(3)))

#if defined(__has_builtin)
#  if __has_builtin(__builtin_amdgcn_global_load_async_to_lds_b32)
#    define HAVE_ASYNC_B32 1
#  endif
#  if __has_builtin(__builtin_amdgcn_s_wait_asynccnt)
#    define HAVE_WAIT_ASYNC 1
#  endif
#endif

#define B_  32
#define CL_ 128
#define C_  256
#define HW_ 4096   // 64*64

__global__ __launch_bounds__(256) void
whitecat_dequant_kernel(const int* __restrict__ left,
                        const int* __restrict__ right,
                        const float* __restrict__ codebook,
                        float* __restrict__ out)
{
    const int t  = threadIdx.x;       // 0..255
    const int bc = blockIdx.x;        // b*C + c
    const int b  = bc >> 8;
    const int c  = bc & 255;

    __shared__ float lut[256];        // 1 KiB codebook row for channel c

    // ---- Stage codebook row c into LDS (async global->LDS, ASYNCcnt) ----
    const float* grow = codebook + (c << 8) + t;   // one f32 per lane
#if defined(HAVE_ASYNC_B32)
    __builtin_amdgcn_global_load_async_to_lds_b32(
        (AS1 int*)(uintptr_t)grow,
        (AS3 int*)(uint32_t)(uintptr_t)&lut[t],
        /*offset=*/0, /*cpol=*/0);
#else
    {
        uint64_t ga = (uint64_t)(uintptr_t)grow;
        uint32_t la = (uint32_t)(uintptr_t)&lut[t];   // LDS byte offset
        asm volatile("global_load_async_to_lds_b32 %0, %1, off"
                     :: "v"(la), "v"(ga)
                     : "memory");
    }
#endif
#if defined(HAVE_WAIT_ASYNC)
    __builtin_amdgcn_s_wait_asynccnt(0);
#else
    asm volatile("s_wait_asynccnt 0" ::: "memory");
#endif
    __syncthreads();   // all 8 waves' LDS writes visible

    // ---- Source plane: left for c<128, right for c>=128 ----
    const int* src = (c < CL_)
        ? (left  + ((size_t)(b * CL_ + c)         << 12))
        : (right + ((size_t)(b * CL_ + (c - CL_)) << 12));
    const v4i* __restrict__ src4 = (const v4i*)src;          // 1024 int4
    v4f* __restrict__       dst4 = (v4f*)(out + ((size_t)bc << 12));

    // 4096 elems / 256 threads = 16 per thread, as 4x int4/float4.
    // Consecutive lanes hit consecutive 16B chunks -> fully coalesced.
#pragma unroll
    for (int i = 0; i < 4; ++i) {
        const int e = t + (i << 8);                 // 0..1023
        v4i codes = __builtin_nontemporal_load(src4 + e);
        v4f r;
        r.x = lut[codes.x & 255];                   // ds_load_b32 gathers
        r.y = lut[codes.y & 255];
        r.z = lut[codes.z & 255];
        r.w = lut[codes.w & 255];
        __builtin_nontemporal_store(r, dst4 + e);   // streaming store (NT)
    }
}

extern "C" void kernel_launch(void* const* d_in, const int* in_sizes, int n_in,
                              void* d_out, int out_size, void* d_ws, size_t ws_size,
                              hipStream_t stream)
{
    const int*   left     = (const int*)d_in[0];
    const int*   right    = (const int*)d_in[1];
    const float* codebook = (const float*)d_in[2];
    float*       out      = (float*)d_out;

    // One block per (b, c) plane: 32 * 256 = 8192 blocks, 256 threads each.
    whitecat_dequant_kernel<<<dim3(B_ * C_), dim3(256), 0, stream>>>(
        left, right, codebook, out);
}